// lightSelfAtten_43688407335539
// MI455X (gfx1250) — compile-verified
//
#include <hip/hip_runtime.h>
#include <hip/hip_bf16.h>
#include <stdint.h>

// Problem constants (match reference)
#define B_BATCH 16
#define C_CH    128
#define H_DIM   128
#define W_DIM   128
#define N_SP    (H_DIM * W_DIM)   // 16384
#define KCH     64                // K-chunk per double-buffer stage

typedef __attribute__((ext_vector_type(2))) float v2f;
typedef __attribute__((ext_vector_type(8))) float v8f;

// ---- CDNA5 async global->LDS helpers (inline asm; ASYNCcnt-tracked) --------
__device__ __forceinline__ void async_ld_b128(uint32_t lds_off, uint32_t voff,
                                              uint64_t sbase) {
    // GLOBAL_LOAD_ASYNC_TO_LDS_B128, GVS mode: saddr(64b SGPR) + vaddr(32b VGPR)
    asm volatile("global_load_async_to_lds_b128 %0, %1, %2"
                 :: "v"(lds_off), "v"(voff), "s"(sbase) : "memory");
}
#define WAIT_ASYNC16() asm volatile("s_wait_asynccnt 0x10" ::: "memory")
#define WAIT_ASYNC0()  asm volatile("s_wait_asynccnt 0x0"  ::: "memory")
#define WAIT_DS0()     asm volatile("s_wait_dscnt 0x0"     ::: "memory")

// -----------------------------------------------------------------------------
// Kernel 1: Gram matrix  G[b] = F F^T  (symmetric: upper-triangle tiles only).
// One wave per WG computes a 16x16 tile; A/B panels (16 x 64 floats) are
// double-buffered in LDS via async global->LDS b128 loads overlapped with the
// 16 f32 WMMAs of the previous chunk.
// -----------------------------------------------------------------------------
__global__ __launch_bounds__(32)
void gram_kernel(const float* __restrict__ x, float* __restrict__ G) {
    const int b = blockIdx.z;
    int t = blockIdx.x;                     // 0..35 -> (tm,tn) with tm <= tn
    int tm = 0;
    while (t >= 8 - tm) { t -= 8 - tm; ++tm; }
    const int tn = tm + t;

    const int lane = threadIdx.x;           // 0..31

    // 68-float row stride: 272B (16B-aligned) and 4*r bank offset -> the 16
    // column reads hit 16 distinct banks; +2 half-wave offset stays disjoint.
    __shared__ float As[2][16][68];
    __shared__ float Bs[2][16][68];

    const float* Fb = x + (size_t)b * C_CH * N_SP;
    const uint64_t aBase = (uint64_t)(uintptr_t)(Fb + (size_t)(tm * 16) * N_SP);
    const uint64_t bBase = (uint64_t)(uintptr_t)(Fb + (size_t)(tn * 16) * N_SP);

    const int lrow = lane >> 4;             // 0/1: sub-row within a row pair
    const int lc   = (lane & 15) * 4;       // float4 column within the chunk

    // Per-lane byte offsets (include buffer 0 base + lane position)
    const uint32_t ldsA0 = (uint32_t)(size_t)&As[0][0][0] + (uint32_t)(lrow * 68 + lc) * 4u;
    const uint32_t ldsB0 = (uint32_t)(size_t)&Bs[0][0][0] + (uint32_t)(lrow * 68 + lc) * 4u;
    const uint32_t bufStride = 16u * 68u * 4u;              // bytes per buffer
    const uint32_t vLane = (uint32_t)(lrow * N_SP + lc) * 4u;

    const int m    = lane & 15;             // A row / B col within tile
    const int koff = (lane >> 4) * 2;       // K-pair select per ISA layout

    v8f acc = {};

    // Issue one chunk: 8 row-pairs x 2 panels = 16 async b128 loads
    auto issue_chunk = [&](uint32_t bufOff, int k0) {
        const uint32_t vb = vLane + (uint32_t)k0 * 4u;
        #pragma unroll
        for (int rr = 0; rr < 8; ++rr) {
            const uint32_t l = bufOff + (uint32_t)rr * (2u * 68u * 4u);
            const uint32_t v = vb + (uint32_t)rr * (2u * N_SP * 4u);
            async_ld_b128(ldsA0 + l, v, aBase);
            async_ld_b128(ldsB0 + l, v, bBase);
        }
    };

    #define GRAM_COMPUTE(BUF)                                                   \
        _Pragma("unroll")                                                       \
        for (int kk = 0; kk < KCH; kk += 4) {                                   \
            v2f a, bf;                                                          \
            a.x  = As[BUF][m][kk + koff];                                       \
            a.y  = As[BUF][m][kk + koff + 1];                                   \
            bf.x = Bs[BUF][m][kk + koff];                                       \
            bf.y = Bs[BUF][m][kk + koff + 1];                                   \
            acc = __builtin_amdgcn_wmma_f32_16x16x4_f32(                        \
                false, a, false, bf, (short)0, acc, false, false);              \
        }

    issue_chunk(0u, 0);                                     // prime buffer 0

    for (int k0 = 0; k0 < N_SP; k0 += 2 * KCH) {
        // Phase 0: prefetch buf1 (k0+KCH), compute buf0 (k0)
        WAIT_DS0();                                         // WAR vs prior reads of buf1
        issue_chunk(bufStride, k0 + KCH);
        WAIT_ASYNC16();                                     // buf0's 16 loads landed
        GRAM_COMPUTE(0)

        // Phase 1: prefetch buf0 (k0+2*KCH) if any, compute buf1 (k0+KCH)
        WAIT_DS0();                                         // WAR vs reads of buf0
        if (k0 + 2 * KCH < N_SP) {
            issue_chunk(0u, k0 + 2 * KCH);
            WAIT_ASYNC16();
        } else {
            WAIT_ASYNC0();
        }
        GRAM_COMPUTE(1)
    }
    #undef GRAM_COMPUTE

    // C/D layout: VGPR j, lanes 0-15 -> M=j, lanes 16-31 -> M=j+8; N = lane&15
    float* Gb = G + (size_t)b * C_CH * C_CH;
    const int mrow = (lane >> 4) * 8;
    const int nn   = lane & 15;
    #pragma unroll
    for (int j = 0; j < 8; ++j)
        Gb[(size_t)(tm * 16 + mrow + j) * C_CH + tn * 16 + nn] = acc[j];
    if (tm != tn) {                                         // mirror: G is symmetric
        #pragma unroll
        for (int j = 0; j < 8; ++j)
            Gb[(size_t)(tn * 16 + nn) * C_CH + tm * 16 + mrow + j] = acc[j];
    }
}

// -----------------------------------------------------------------------------
// Kernel 2: A = rowmax(G) - G, in place. One 128-thread block per row.
// -----------------------------------------------------------------------------
__global__ __launch_bounds__(128)
void rowmax_kernel(float* __restrict__ G) {
    const int row = blockIdx.x;            // b*128 + c  (2048 rows)
    const int t   = threadIdx.x;           // 0..127
    float* g = G + (size_t)row * C_CH;

    float v = g[t];
    float m = v;
    #pragma unroll
    for (int off = 16; off > 0; off >>= 1)
        m = fmaxf(m, __shfl_xor(m, off, 32));

    __shared__ float wmax[4];
    if ((t & 31) == 0) wmax[t >> 5] = m;
    __syncthreads();
    m = fmaxf(fmaxf(wmax[0], wmax[1]), fmaxf(wmax[2], wmax[3]));

    g[t] = m - v;
}

// -----------------------------------------------------------------------------
// Kernel 3: Y[b] = A[b] (128x128) @ F[b] (128x16384)
// 8 waves per WG share the 16x128 A-panel in LDS; each wave computes one
// 16x16 output tile (K=128 -> 32 WMMAs). B fragments loaded from global
// (two 64B-contiguous segments per load, L2-resident).
// -----------------------------------------------------------------------------
__global__ __launch_bounds__(256)
void av_kernel(const float* __restrict__ x, const float* __restrict__ A,
               float* __restrict__ Y) {
    const int b    = blockIdx.z;
    const int tm   = blockIdx.y;                       // 0..7
    const int wave = threadIdx.x >> 5;
    const int lane = threadIdx.x & 31;
    const int n0   = blockIdx.x * 128 + wave * 16;     // 128 n-chunks

    __shared__ float As[16][132];
    const float* Ab = A + ((size_t)b * C_CH + tm * 16) * C_CH;
    for (int i = threadIdx.x; i < 512; i += 256) {     // 512 float4 = 16x128 floats
        const int r  = i >> 5;
        const int c4 = (i & 31) * 4;
        *(float4*)&As[r][c4] = *(const float4*)(Ab + (size_t)r * C_CH + c4);
    }
    __syncthreads();

    const float* Fb = x + (size_t)b * C_CH * N_SP;
    const int m     = lane & 15;
    const int koff  = (lane >> 4) * 2;

    v8f acc = {};
    #pragma unroll 4
    for (int kk = 0; kk < 128; kk += 4) {
        v2f a, bf;
        a.x = As[m][kk + koff];
        a.y = As[m][kk + koff + 1];
        const float* Bp = Fb + (size_t)(kk + koff) * N_SP + n0 + m;
        bf.x = Bp[0];
        bf.y = Bp[N_SP];
        acc = __builtin_amdgcn_wmma_f32_16x16x4_f32(
            false, a, false, bf, (short)0, acc, false, false);
    }

    float* Yt = Y + ((size_t)b * C_CH + tm * 16) * N_SP + n0;
    const int mrow = (lane >> 4) * 8;
    #pragma unroll
    for (int j = 0; j < 8; ++j)
        Yt[(size_t)(mrow + j) * N_SP + (lane & 15)] = acc[j];
}

// -----------------------------------------------------------------------------
// Kernel 4: depthwise 3x3 conv (pad=1) + bias.  One channel image per WG.
// Reads Y (atten half of d_out), writes Z (out half) -- disjoint buffers.
// -----------------------------------------------------------------------------
__global__ __launch_bounds__(256)
void dwconv_kernel(const float* __restrict__ Yin, const float* __restrict__ wgt,
                   const float* __restrict__ bias, float* __restrict__ Z) {
    const int bc = blockIdx.x;             // b*128 + c
    const int c  = bc & (C_CH - 1);
    const float* img  = Yin + (size_t)bc * N_SP;
    float*       zimg = Z   + (size_t)bc * N_SP;

    float w[9];
    #pragma unroll
    for (int i = 0; i < 9; ++i) w[i] = wgt[c * 9 + i];
    const float bs = bias[c];

    for (int p = threadIdx.x; p < N_SP; p += 256) {
        const int h  = p >> 7;
        const int wc = p & 127;
        float acc = bs;
        #pragma unroll
        for (int dh = -1; dh <= 1; ++dh) {
            const int hh = h + dh;
            if (hh < 0 || hh > H_DIM - 1) continue;
            #pragma unroll
            for (int dw = -1; dw <= 1; ++dw) {
                const int ww = wc + dw;
                if (ww < 0 || ww > W_DIM - 1) continue;
                acc += w[(dh + 1) * 3 + (dw + 1)] * img[hh * W_DIM + ww];
            }
        }
        zimg[p] = acc;
    }
}

// -----------------------------------------------------------------------------
// Kernel 5: softmax over W (row of 128) + out = gamma*atten + x.
// One row per wave, 4 cols per lane. Reads Z from out-half (row-local),
// writes atten (atten half) and out (out half, overwriting Z it just read).
// -----------------------------------------------------------------------------
__global__ __launch_bounds__(256)
void softmax_combine_kernel(const float* Z, const float* __restrict__ x,
                            const float* __restrict__ gamma,
                            float* out, float* __restrict__ atten) {
    const int wave   = threadIdx.x >> 5;
    const int lane   = threadIdx.x & 31;
    const size_t row = (size_t)blockIdx.x * 8 + wave;   // 262144 rows total
    const size_t off = row * W_DIM + lane * 4;
    const float  g   = gamma[0];

    float4 v = *(const float4*)(Z + off);
    float m = fmaxf(fmaxf(v.x, v.y), fmaxf(v.z, v.w));
    #pragma unroll
    for (int o = 16; o > 0; o >>= 1) m = fmaxf(m, __shfl_xor(m, o, 32));

    float4 e;
    e.x = __expf(v.x - m); e.y = __expf(v.y - m);
    e.z = __expf(v.z - m); e.w = __expf(v.w - m);
    float s = e.x + e.y + e.z + e.w;
    #pragma unroll
    for (int o = 16; o > 0; o >>= 1) s += __shfl_xor(s, o, 32);
    const float inv = 1.0f / s;

    float4 a;
    a.x = e.x * inv; a.y = e.y * inv; a.z = e.z * inv; a.w = e.w * inv;
    *(float4*)(atten + off) = a;

    const float4 xv = *(const float4*)(x + off);
    float4 o4;
    o4.x = g * a.x + xv.x; o4.y = g * a.y + xv.y;
    o4.z = g * a.z + xv.z; o4.w = g * a.w + xv.w;
    *(float4*)(out + off) = o4;
}

// -----------------------------------------------------------------------------
extern "C" void kernel_launch(void* const* d_in, const int* in_sizes, int n_in,
                              void* d_out, int out_size, void* d_ws, size_t ws_size,
                              hipStream_t stream) {
    (void)in_sizes; (void)n_in; (void)out_size; (void)ws_size;

    const float* x     = (const float*)d_in[0];   // [B,C,H,W]
    const float* dww   = (const float*)d_in[1];   // [C,1,3,3]
    const float* dwb   = (const float*)d_in[2];   // [C]
    const float* gamma = (const float*)d_in[3];   // [1]

    float* out   = (float*)d_out;                                   // out half
    float* atten = (float*)d_out + (size_t)B_BATCH * C_CH * N_SP;   // atten half (Y scratch)
    float* G     = (float*)d_ws;                                    // [B,C,C] = 1 MB

    // 1) Gram matrix G = F F^T  (WMMA f32, async-LDS double buffered, symmetric)
    dim3 g1(36, 1, B_BATCH);
    gram_kernel<<<g1, 32, 0, stream>>>(x, G);

    // 2) A = rowmax(G) - G  (in place)
    rowmax_kernel<<<B_BATCH * C_CH, 128, 0, stream>>>(G);

    // 3) Y = A @ F  -> atten half as scratch  (WMMA f32)
    dim3 g3(N_SP / 128, 8, B_BATCH);
    av_kernel<<<g3, 256, 0, stream>>>(x, G, atten);

    // 4) depthwise conv + bias: Y -> Z (out half)
    dwconv_kernel<<<B_BATCH * C_CH, 256, 0, stream>>>(atten, dww, dwb, out);

    // 5) softmax over W + out = gamma*atten + x
    softmax_combine_kernel<<<(B_BATCH * C_CH * H_DIM) / 8, 256, 0, stream>>>(
        out, x, gamma, out, atten);
}